// DPConv_39771397161537
// MI455X (gfx1250) — compile-verified
//
#include <hip/hip_runtime.h>
#include <hip/hip_bf16.h>

// ---------------------------------------------------------------------------
// DPConv pyramid-patch kernel for MI455X (gfx1250, wave32).
//
// Pass 1: y[n,o,h,w] = sum_c x[n,c,h',w'] * w_reduce[o,c] + b[o]   (edge pad)
//   implemented with V_WMMA_F32_16X16X4_F32: M=16 out-channels,
//   N=16 pixels per wave, K=64 channels in 16 steps of 4.
// Pass 2: gather/resample y -> out for the three patch pyramids.
//   Bandwidth-bound: y (34 MB) fits in the 192 MB L2, out writes coalesced.
// ---------------------------------------------------------------------------

typedef __attribute__((ext_vector_type(2))) float v2f;
typedef __attribute__((ext_vector_type(8))) float v8f;

#define HP   258                   // padded height/width
#define HW   (HP * HP)             // 66564 pixels per (n, channel) plane
#define CIN  64
#define COUT 16
#define NB   8
#define TOTPIX ((long)NB * HW)     // 532512, exactly divisible by 16
#define NTILES (TOTPIX / 16)       // 33282 waves of work

__global__ __launch_bounds__(256)
void dpconv_reduce_wmma(const float* __restrict__ x,
                        const float* __restrict__ wr,
                        const float* __restrict__ br,
                        float* __restrict__ y)
{
    const int lane = threadIdx.x & 31;
    const long waveId = (long)blockIdx.x * 8 + (threadIdx.x >> 5);
    if (waveId >= NTILES) return;          // wave-uniform: EXEC stays all-1s

    const int half = lane >> 4;            // 0: K={0,1}, 1: K={2,3} slice
    const int nIdx = lane & 15;            // N index (pixel) == A-matrix row M

    // Decode this lane's pixel of the padded plane.
    long pix = waveId * 16 + nIdx;
    int n  = (int)(pix / HW);
    int hw = (int)(pix % HW);
    int h  = hw / HP;
    int w  = hw % HP;
    int hs = min(max(h - 1, 0), 255);      // edge-pad clamp into source
    int wsrc = min(max(w - 1, 0), 255);

    const float* xb = x + ((long)n * CIN) * 65536 + (long)hs * 256 + wsrc;
    const float* wb = wr + nIdx * CIN + half * 2;   // A row = nIdx, K offset

    v8f acc = {};
    #pragma unroll
    for (int k0 = 0; k0 < CIN; k0 += 4) {
        v2f A, B;
        // A (16x4 f32): lanes 0-15 -> M=0..15 K={k0,k0+1};
        //               lanes16-31 -> M=0..15 K={k0+2,k0+3}
        A.x = wb[k0 + 0];
        A.y = wb[k0 + 1];
        // B (4x16 f32): same K split across lane halves, N=pixel across lanes
        int c0 = k0 + half * 2;
        B.x = xb[(long)(c0 + 0) * 65536];
        B.y = xb[(long)(c0 + 1) * 65536];
        acc = __builtin_amdgcn_wmma_f32_16x16x4_f32(
            /*neg_a=*/false, A, /*neg_b=*/false, B,
            /*c_mod=*/(short)0, acc, /*reuse_a=*/false, /*reuse_b=*/false);
    }

    // D layout: lanes 0-15: VGPR j -> (M=j,   N=lane);
    //           lanes16-31: VGPR j -> (M=j+8, N=lane-16)
    float* yb = y + ((long)n * COUT) * HW + (long)h * HP + w;
    #pragma unroll
    for (int j = 0; j < 8; ++j) {
        int o = j + half * 8;
        yb[(long)o * HW] = acc[j] + br[o];
    }
}

// ---------------------------------------------------------------------------
// Pass 2: one thread per output element of (8, 11909, 16, 8, 8).
//   l <  4096          : k=4  patches, bilinear 4x4 -> 8x8
//   l <  4096+3969     : k=8  patches, identity copy
//   else               : k=12 patches, adaptive avg (2x2 mean, weight 0.25)
// ---------------------------------------------------------------------------
#define L0 4096      // 64*64
#define L1 3969      // 63*63
#define L2SEG 3844   // 62*62
#define LTOT (L0 + L1 + L2SEG)   // 11909

__global__ __launch_bounds__(256)
void dpconv_resample(const float* __restrict__ y,
                     float* __restrict__ out,
                     long total)
{
    long idx = (long)blockIdx.x * blockDim.x + threadIdx.x;
    if (idx >= total) return;

    int within = (int)(idx & 1023);        // c*64 + p*8 + q
    long nl = idx >> 10;
    int c = within >> 6;
    int p = (within >> 3) & 7;
    int q = within & 7;
    int n = (int)(nl / LTOT);
    int l = (int)(nl % LTOT);

    const float* yb = y + ((long)(n * COUT + c)) * HW;
    float val;

    if (l < L0) {
        // ---- k=4, bilinear upsample (half-pixel centers, edge clamp) ----
        int oh = l >> 6, ow = l & 63;
        int top = oh * 4, left = ow * 4;
        float sp = 0.5f * (float)p - 0.25f;
        float sq = 0.5f * (float)q - 0.25f;
        int ip = (int)floorf(sp); float tp = sp - (float)ip;
        int iq = (int)floorf(sq); float tq = sq - (float)iq;
        int p0 = min(max(ip, 0), 3), p1 = min(max(ip + 1, 0), 3);
        int q0 = min(max(iq, 0), 3), q1 = min(max(iq + 1, 0), 3);
        const float* r0 = yb + (long)(top + p0) * HP + left;
        const float* r1 = yb + (long)(top + p1) * HP + left;
        float v00 = r0[q0], v01 = r0[q1];
        float v10 = r1[q0], v11 = r1[q1];
        val = (1.0f - tp) * ((1.0f - tq) * v00 + tq * v01)
            +         tp  * ((1.0f - tq) * v10 + tq * v11);
    } else if (l < L0 + L1) {
        // ---- k=8, identity ----
        int l2 = l - L0;
        int oh = l2 / 63, ow = l2 % 63;
        val = yb[(long)(oh * 4 + p) * HP + (ow * 4 + q)];
    } else {
        // ---- k=12, adaptive avg: rows (3p)>>1..+1, cols (3q)>>1..+1 ----
        int l3 = l - (L0 + L1);
        int oh = l3 / 62, ow = l3 % 62;
        int rp = (3 * p) >> 1, cq = (3 * q) >> 1;
        const float* r0 = yb + (long)(oh * 4 + rp) * HP + (ow * 4 + cq);
        const float* r1 = r0 + HP;
        val = 0.25f * (r0[0] + r0[1] + r1[0] + r1[1]);
    }

    out[idx] = val;
}

extern "C" void kernel_launch(void* const* d_in, const int* in_sizes, int n_in,
                              void* d_out, int out_size, void* d_ws, size_t ws_size,
                              hipStream_t stream) {
    const float* x  = (const float*)d_in[0];   // (8,64,256,256)
    const float* wr = (const float*)d_in[1];   // (16,64)
    const float* br = (const float*)d_in[2];   // (16,)
    float* y = (float*)d_ws;                   // needs 8*16*258*258*4 = 34 MB

    // Pass 1: WMMA channel reduction into workspace.
    int blocks1 = (int)((NTILES + 7) / 8);     // 8 waves (256 thr) per block
    dpconv_reduce_wmma<<<blocks1, 256, 0, stream>>>(x, wr, br, y);

    // Pass 2: resample/gather into d_out.
    long total = (long)out_size;               // 97,558,528 floats
    long blocks2 = (total + 255) / 256;
    dpconv_resample<<<(unsigned)blocks2, 256, 0, stream>>>(y, (float*)d_out, total);
}